// MHIMModel_23398981829298
// MI455X (gfx1250) — compile-verified
//
#include <hip/hip_runtime.h>
#include <hip/hip_bf16.h>

typedef __attribute__((ext_vector_type(2))) float v2f;
typedef __attribute__((ext_vector_type(8))) float v8f;

constexpr int kN  = 50000;   // nodes per modality
constexpr int kM  = 20000;   // hyperedges per modality
constexpr int kE  = 200000;  // incidences per modality
constexpr int kD  = 128;     // embedding dim
constexpr int kH  = 8;       // heads
constexpr int kC  = 64;      // context entities
constexpr int kNE = 200000;  // n_entity
constexpr int kR  = 3 * kN;  // 150000 rows of `related`
constexpr int kHD = kD / kH; // 16 head dim
constexpr int kCHUNK  = 512;
constexpr int kTILE   = 128;
constexpr int kNCHUNK = (kR + kCHUNK - 1) / kCHUNK;  // 293
constexpr int kPSTRIDE = kHD + 2;                    // m, l, acc[16]

// ---------------------------------------------------------------------------
// Utility: zero a float buffer
// ---------------------------------------------------------------------------
__global__ void zero_kernel(float* __restrict__ p, long n) {
  long i = (long)blockIdx.x * blockDim.x + threadIdx.x;
  if (i < n) p[i] = 0.0f;
}

// ---------------------------------------------------------------------------
// Dense GEMM: Out[rows,128] = A[rows,128] @ W[128,128] (+ bias per column)
// One 16-row strip per block; 8 waves, each wave owns one 16-col tile and
// runs 32 chained V_WMMA_F32_16X16X4_F32 ops over K=128.
// A-frag (16x4 f32): lane L holds row M=L%16; VGPR0/1 hold K = 2*(L/16)+{0,1}.
// B-frag (4x16 f32): lane L holds col N=L%16; VGPR0/1 hold K = 2*(L/16)+{0,1}.
// C/D (16x16 f32):   VGPR r -> row r + 8*(L/16), col = L%16.
// ---------------------------------------------------------------------------
__global__ void gemm128_wmma(const float* __restrict__ A,
                             const float* __restrict__ W,
                             const float* __restrict__ bias,
                             float* __restrict__ Out) {
  const int m0    = blockIdx.x * 16;
  const int wave  = threadIdx.x >> 5;   // 0..7
  const int lane  = threadIdx.x & 31;
  const int n0    = wave * 16;
  const int mrow  = lane & 15;
  const int khalf = (lane >> 4) << 1;   // 0 or 2

  v8f c = 0.0f;
  const float* arow = A + (size_t)(m0 + mrow) * kD;
  for (int k = 0; k < kD; k += 4) {
    v2f a, b;
    a.x = arow[k + khalf];
    a.y = arow[k + khalf + 1];
    b.x = W[(size_t)(k + khalf) * kD + n0 + mrow];
    b.y = W[(size_t)(k + khalf + 1) * kD + n0 + mrow];
    c = __builtin_amdgcn_wmma_f32_16x16x4_f32(false, a, false, b,
                                              (short)0, c, false, false);
  }
  const int rbase = m0 + ((lane >> 4) << 3);
  const int col   = n0 + mrow;
  const float bv  = bias ? bias[col] : 0.0f;
#pragma unroll
  for (int r = 0; r < 8; ++r)
    Out[(size_t)(rbase + r) * kD + col] = c[r] + bv;
}

// ---------------------------------------------------------------------------
// Hypergraph conv helpers
// ---------------------------------------------------------------------------
__global__ void degree_kernel(const int* __restrict__ node,
                              const int* __restrict__ edge,
                              float* __restrict__ degN,
                              float* __restrict__ degE, int ecount) {
  int i = blockIdx.x * blockDim.x + threadIdx.x;
  if (i < ecount) {
    atomicAdd(&degE[edge[i]], 1.0f);
    atomicAdd(&degN[node[i]], 1.0f);
  }
}

// e[edge[i],:] += xt[node[i],:]   (32 threads per incidence, float4 granule)
__global__ void scatter_edge_kernel(const float* __restrict__ xt,
                                    const int* __restrict__ node,
                                    const int* __restrict__ edge,
                                    float* __restrict__ ebuf, int ecount) {
  int t = blockIdx.x * blockDim.x + threadIdx.x;
  if (t >= ecount * 32) return;
  int i = t >> 5, sub = t & 31;
  int n = node[i], m = edge[i];
  float4 v = reinterpret_cast<const float4*>(xt + (size_t)n * kD)[sub];
  float* dst = ebuf + (size_t)m * kD + sub * 4;
  atomicAdd(dst + 0, v.x);
  atomicAdd(dst + 1, v.y);
  atomicAdd(dst + 2, v.z);
  atomicAdd(dst + 3, v.w);
}

__global__ void scale_edge_kernel(float* __restrict__ ebuf,
                                  const float* __restrict__ degE, long n) {
  long t = (long)blockIdx.x * blockDim.x + threadIdx.x;
  if (t < n) {
    float d = degE[t >> 7];
    ebuf[t] *= (d > 0.0f) ? (1.0f / d) : 0.0f;
  }
}

// out[node[i],:] += e[edge[i],:]
__global__ void scatter_node_kernel(const float* __restrict__ ebuf,
                                    const int* __restrict__ node,
                                    const int* __restrict__ edge,
                                    float* __restrict__ out, int ecount) {
  int t = blockIdx.x * blockDim.x + threadIdx.x;
  if (t >= ecount * 32) return;
  int i = t >> 5, sub = t & 31;
  int n = node[i], m = edge[i];
  float4 v = reinterpret_cast<const float4*>(ebuf + (size_t)m * kD)[sub];
  float* dst = out + (size_t)n * kD + sub * 4;
  atomicAdd(dst + 0, v.x);
  atomicAdd(dst + 1, v.y);
  atomicAdd(dst + 2, v.z);
  atomicAdd(dst + 3, v.w);
}

__global__ void scale_node_kernel(float* __restrict__ out,
                                  const float* __restrict__ degN,
                                  const float* __restrict__ bias, long n) {
  long t = (long)blockIdx.x * blockDim.x + threadIdx.x;
  if (t < n) {
    float d = degN[t >> 7];
    out[t] = out[t] * ((d > 0.0f) ? (1.0f / d) : 0.0f) + bias[t & (kD - 1)];
  }
}

// ---------------------------------------------------------------------------
// Flash attention over R=150000 rows. Block = (head h, chunk ch of 512 rows),
// 256 threads = 8 waves. K/V for each 128-row tile are projected with WMMA:
// each wave owns a 16-row sub-tile and runs 2x32 chained
// V_WMMA_F32_16X16X4_F32 ops ([16x128] @ [128x16], B-frag from LDS-resident
// per-head weight slices). Online softmax per query uses wave32 shuffle
// butterflies. Partials (m, l, acc[16]) per (h, chunk, query) go to global
// for a combine pass.
// ---------------------------------------------------------------------------
__global__ void attn_kernel(const float* __restrict__ related,
                            const float* __restrict__ wk, const float* __restrict__ bk,
                            const float* __restrict__ wv, const float* __restrict__ bv,
                            const float* __restrict__ Q,
                            float* __restrict__ partials) {
  __shared__ float sWk[kD][kHD];
  __shared__ float sWv[kD][kHD];
  __shared__ float sQ[kC][kHD];
  __shared__ float sK[kTILE][kHD];
  __shared__ float sV[kTILE][kHD];
  __shared__ float sM[kC];
  __shared__ float sL[kC];
  __shared__ float sAcc[kC][kHD];
  __shared__ float sbk[kHD], sbv[kHD];

  const int h    = blockIdx.x;
  const int ch   = blockIdx.y;
  const int tid  = threadIdx.x;
  const int lane = tid & 31;
  const int wid  = tid >> 5;

  for (int idx = tid; idx < kD * kHD; idx += 256) {
    int d = idx >> 4, j = idx & 15;
    sWk[d][j] = wk[d * kD + h * kHD + j];
    sWv[d][j] = wv[d * kD + h * kHD + j];
  }
  for (int idx = tid; idx < kC * kHD; idx += 256) {
    int c = idx >> 4, j = idx & 15;
    sQ[c][j]   = Q[c * kD + h * kHD + j] * 0.25f;  // 1/sqrt(hd)=0.25
    sAcc[c][j] = 0.0f;
  }
  if (tid < kC) { sM[tid] = -__builtin_inff(); sL[tid] = 0.0f; }
  if (tid < kHD) { sbk[tid] = bk[h * kHD + tid]; sbv[tid] = bv[h * kHD + tid]; }
  __syncthreads();

  const int mrow  = lane & 15;
  const int khalf = (lane >> 4) << 1;   // 0 or 2

  const int chunk_start = ch * kCHUNK;
  for (int t = 0; t < kCHUNK / kTILE; ++t) {
    const int tile_start = chunk_start + t * kTILE;
    int tile_valid = kR - tile_start;
    if (tile_valid > kTILE) tile_valid = kTILE;
    if (tile_valid <= 0) break;  // uniform across the block

    // --- WMMA K/V projection: wave `wid` owns rows [wid*16, wid*16+16) ---
    {
      int rglob = tile_start + wid * 16 + mrow;
      if (rglob >= kR) rglob = kR - 1;  // clamp (masked to p=0 in softmax)
      const float* arow = related + (size_t)rglob * kD;
      // prefetch next tile's row for this lane (global_prefetch_b8)
      {
        int rnext = rglob + kTILE;
        if (rnext >= kR) rnext = kR - 1;
        __builtin_prefetch(related + (size_t)rnext * kD, 0, 1);
      }
      v8f ck = 0.0f, cv = 0.0f;
      for (int k = 0; k < kD; k += 4) {
        v2f a, bkf, bvf;
        a.x   = arow[k + khalf];
        a.y   = arow[k + khalf + 1];
        bkf.x = sWk[k + khalf][mrow];
        bkf.y = sWk[k + khalf + 1][mrow];
        bvf.x = sWv[k + khalf][mrow];
        bvf.y = sWv[k + khalf + 1][mrow];
        ck = __builtin_amdgcn_wmma_f32_16x16x4_f32(false, a, false, bkf,
                                                   (short)0, ck, false, false);
        cv = __builtin_amdgcn_wmma_f32_16x16x4_f32(false, a, false, bvf,
                                                   (short)0, cv, false, false);
      }
      const int rbase = wid * 16 + ((lane >> 4) << 3);
#pragma unroll
      for (int r = 0; r < 8; ++r) {
        sK[rbase + r][mrow] = ck[r] + sbk[mrow];
        sV[rbase + r][mrow] = cv[r] + sbv[mrow];
      }
    }
    __syncthreads();

    // --- scores + online softmax: wave `wid` owns queries wid*8 .. wid*8+7 ---
    for (int qi = 0; qi < 8; ++qi) {
      int c = wid * 8 + qi;
      float s[4];
#pragma unroll
      for (int kk = 0; kk < 4; ++kk) {
        int rl = lane + kk * 32;
        if (rl < tile_valid) {
          float acc = 0.0f;
#pragma unroll
          for (int j = 0; j < kHD; ++j) acc += sQ[c][j] * sK[rl][j];
          s[kk] = acc;
        } else {
          s[kk] = -__builtin_inff();
        }
      }
      float tmax = fmaxf(fmaxf(s[0], s[1]), fmaxf(s[2], s[3]));
      for (int off = 16; off > 0; off >>= 1)
        tmax = fmaxf(tmax, __shfl_xor(tmax, off, 32));
      float mold = sM[c];
      float mnew = fmaxf(mold, tmax);
      float p[4], ls = 0.0f;
#pragma unroll
      for (int kk = 0; kk < 4; ++kk) { p[kk] = __expf(s[kk] - mnew); ls += p[kk]; }
      for (int off = 16; off > 0; off >>= 1) ls += __shfl_xor(ls, off, 32);
      float part[kHD];
#pragma unroll
      for (int j = 0; j < kHD; ++j)
        part[j] = p[0] * sV[lane][j] + p[1] * sV[lane + 32][j] +
                  p[2] * sV[lane + 64][j] + p[3] * sV[lane + 96][j];
#pragma unroll
      for (int j = 0; j < kHD; ++j)
        for (int off = 16; off > 0; off >>= 1)
          part[j] += __shfl_xor(part[j], off, 32);
      float f = __expf(mold - mnew);
      if (lane == 0) {
        sM[c] = mnew;
        sL[c] = sL[c] * f + ls;
#pragma unroll
        for (int j = 0; j < kHD; ++j) sAcc[c][j] = sAcc[c][j] * f + part[j];
      }
    }
    __syncthreads();
  }

  // write partials for this (head, chunk)
  float* pb = partials + ((size_t)h * kNCHUNK + ch) * kC * kPSTRIDE;
  for (int qi = 0; qi < 8; ++qi) {
    int c = wid * 8 + qi;
    float* b = pb + (size_t)c * kPSTRIDE;
    if (lane == 0) { b[0] = sM[c]; b[1] = sL[c]; }
    if (lane < kHD) b[2 + lane] = sAcc[c][lane];
  }
}

// Combine per-chunk softmax partials: one thread per (h, c, j).
__global__ void attn_combine(const float* __restrict__ partials,
                             float* __restrict__ attnH) {
  int t = blockIdx.x * blockDim.x + threadIdx.x;
  if (t >= kH * kC * kHD) return;
  int h   = t / (kC * kHD);
  int rem = t % (kC * kHD);
  int c   = rem / kHD;
  int j   = rem % kHD;
  float M = -__builtin_inff();
  for (int ch = 0; ch < kNCHUNK; ++ch)
    M = fmaxf(M, partials[(((size_t)h * kNCHUNK + ch) * kC + c) * kPSTRIDE]);
  float Ls = 0.0f, O = 0.0f;
  for (int ch = 0; ch < kNCHUNK; ++ch) {
    const float* b = partials + (((size_t)h * kNCHUNK + ch) * kC + c) * kPSTRIDE;
    float w = __expf(b[0] - M);
    Ls += b[1] * w;
    O  += b[2 + j] * w;
  }
  attnH[c * kD + h * kHD + j] = O / Ls;  // concat-heads layout [C, D]
}

// user_repr = (sum_c context + mean_c attentive) / (C+1)
__global__ void user_kernel(const float* __restrict__ context,
                            const float* __restrict__ att,
                            float* __restrict__ u) {
  int d = threadIdx.x;  // 128 threads
  float sc = 0.0f, sa = 0.0f;
  for (int c = 0; c < kC; ++c) { sc += context[c * kD + d]; sa += att[c * kD + d]; }
  u[d] = (sc + sa * (1.0f / kC)) * (1.0f / (kC + 1));
}

// scores[t] = sum_d u[d] * w_rec[d*NE + t] + b_rec[t]   (bandwidth bound)
__global__ void rec_kernel(const float* __restrict__ u,
                           const float* __restrict__ w_rec,
                           const float* __restrict__ b_rec,
                           float* __restrict__ out) {
  __shared__ float su[kD];
  int tid = threadIdx.x;
  if (tid < kD) su[tid] = u[tid];
  __syncthreads();
  int t = blockIdx.x * 256 + tid;
  if (t >= kNE) return;
  float acc = b_rec[t];
#pragma unroll 4
  for (int d = 0; d < kD; ++d) acc += su[d] * w_rec[(size_t)d * kNE + t];
  out[t] = acc;
}

// ---------------------------------------------------------------------------
extern "C" void kernel_launch(void* const* d_in, const int* in_sizes, int n_in,
                              void* d_out, int out_size, void* d_ws, size_t ws_size,
                              hipStream_t stream) {
  (void)in_sizes; (void)n_in; (void)out_size; (void)ws_size;
  const float* x_item     = (const float*)d_in[0];
  const float* theta_item = (const float*)d_in[1];
  const float* bias_item  = (const float*)d_in[2];
  const float* x_ent      = (const float*)d_in[3];
  const float* theta_ent  = (const float*)d_in[4];
  const float* bias_ent   = (const float*)d_in[5];
  const float* x_word     = (const float*)d_in[6];
  const float* theta_word = (const float*)d_in[7];
  const float* bias_word  = (const float*)d_in[8];
  const float* context    = (const float*)d_in[9];
  const float* wq = (const float*)d_in[10];
  const float* bq = (const float*)d_in[11];
  const float* wk = (const float*)d_in[12];
  const float* bk = (const float*)d_in[13];
  const float* wv = (const float*)d_in[14];
  const float* bv = (const float*)d_in[15];
  const float* wo = (const float*)d_in[16];
  const float* bo = (const float*)d_in[17];
  const float* w_rec = (const float*)d_in[18];
  const float* b_rec = (const float*)d_in[19];
  const int* node_item = (const int*)d_in[20];
  const int* edge_item = (const int*)d_in[21];
  const int* node_ent  = (const int*)d_in[22];
  const int* edge_ent  = (const int*)d_in[23];
  const int* node_word = (const int*)d_in[24];
  const int* edge_word = (const int*)d_in[25];
  float* out = (float*)d_out;

  // workspace layout (floats)
  float* ws    = (float*)d_ws;
  float* rel   = ws;                              // 3N*D = 19,200,000
  float* xt    = rel + (size_t)3 * kN * kD;       // N*D
  float* ebuf  = xt + (size_t)kN * kD;            // M*D
  float* degE  = ebuf + (size_t)kM * kD;          // M
  float* degN  = degE + kM;                       // N
  float* Q     = degN + kN;                       // C*D
  float* attnH = Q + (size_t)kC * kD;             // C*D
  float* att   = attnH + (size_t)kC * kD;         // C*D
  float* u     = att + (size_t)kC * kD;           // D
  float* part  = u + kD;                          // H*NCHUNK*C*18

  auto cdiv = [](long a, long b) -> unsigned { return (unsigned)((a + b - 1) / b); };

  zero_kernel<<<cdiv(3L * kN * kD, 256), 256, 0, stream>>>(rel, 3L * kN * kD);

  const float* xs[3] = {x_item, x_ent, x_word};
  const float* th[3] = {theta_item, theta_ent, theta_word};
  const float* bs[3] = {bias_item, bias_ent, bias_word};
  const int* nodes[3] = {node_item, node_ent, node_word};
  const int* edges[3] = {edge_item, edge_ent, edge_word};

  for (int m = 0; m < 3; ++m) {
    gemm128_wmma<<<kN / 16, 256, 0, stream>>>(xs[m], th[m], nullptr, xt);
    zero_kernel<<<cdiv((long)kM * kD, 256), 256, 0, stream>>>(ebuf, (long)kM * kD);
    zero_kernel<<<cdiv(kM, 256), 256, 0, stream>>>(degE, kM);
    zero_kernel<<<cdiv(kN, 256), 256, 0, stream>>>(degN, kN);
    degree_kernel<<<cdiv(kE, 256), 256, 0, stream>>>(nodes[m], edges[m], degN, degE, kE);
    scatter_edge_kernel<<<cdiv((long)kE * 32, 256), 256, 0, stream>>>(xt, nodes[m], edges[m], ebuf, kE);
    scale_edge_kernel<<<cdiv((long)kM * kD, 256), 256, 0, stream>>>(ebuf, degE, (long)kM * kD);
    float* outm = rel + (size_t)m * kN * kD;
    scatter_node_kernel<<<cdiv((long)kE * 32, 256), 256, 0, stream>>>(ebuf, nodes[m], edges[m], outm, kE);
    scale_node_kernel<<<cdiv((long)kN * kD, 256), 256, 0, stream>>>(outm, degN, bs[m], (long)kN * kD);
  }

  gemm128_wmma<<<kC / 16, 256, 0, stream>>>(context, wq, bq, Q);
  attn_kernel<<<dim3(kH, kNCHUNK), 256, 0, stream>>>(rel, wk, bk, wv, bv, Q, part);
  attn_combine<<<cdiv(kH * kC * kHD, 256), 256, 0, stream>>>(part, attnH);
  gemm128_wmma<<<kC / 16, 256, 0, stream>>>(attnH, wo, bo, att);
  user_kernel<<<1, 128, 0, stream>>>(context, att, u);
  rec_kernel<<<cdiv(kNE, 256), 256, 0, stream>>>(u, w_rec, b_rec, out);
}